// TreeEncoder_GATMini_30253749633402
// MI455X (gfx1250) — compile-verified
//
#include <hip/hip_runtime.h>
#include <math.h>
#include <stdint.h>

// ---------------------------------------------------------------------------
// GAT 2-layer forward for MI455X (gfx1250, wave32, WMMA + TDM).
// - GEMMs: V_WMMA_F32_16X16X4_F32, W staged into LDS via TENSOR_LOAD_TO_LDS
//   (Tensor Data Mover) once per workgroup, B-fragments served from LDS.
// - Edge stage dominates (~820 MB L2 traffic/layer, working set fits 192MB L2);
//   GEMMs are ~3.3 GFLOP total, so full f32 precision throughout.
// ---------------------------------------------------------------------------

#define N_NODES 50000
#define N_EDGES 800000
#define IN_DIM  128
#define HIDDEN  16
#define HEADS1  8
#define C1      128   // HEADS1*HIDDEN
#define OUT_DIM 128
#define NEG_SLOPE 0.2f
#define GAT_EPS 1e-16f

typedef __attribute__((ext_vector_type(2))) float v2f;
typedef __attribute__((ext_vector_type(8))) float v8f;
typedef __attribute__((ext_vector_type(4))) unsigned int v4u;
typedef __attribute__((ext_vector_type(8))) int v8i_t;
typedef __attribute__((ext_vector_type(4))) int v4i_t;

#if __has_builtin(__builtin_amdgcn_tensor_load_to_lds)
#define GAT_HAS_TDM 1
#else
#define GAT_HAS_TDM 0
#endif

// ---------------- sign-aware float atomic max (lowers to native int atomics)
__device__ __forceinline__ void atomicMaxF(float* addr, float val) {
    if (val >= 0.0f)
        atomicMax((int*)addr, __float_as_int(val));
    else
        atomicMin((unsigned int*)addr, __float_as_uint(val));
}

// ---------------- WMMA f32 GEMM: C[M,128] = A[M,128] * W[128,128]
// One wave per 16x16 output tile; W staged to LDS by the Tensor Data Mover.
__global__ void __launch_bounds__(128)
gemm16_wmma_f32(const float* __restrict__ A, const float* __restrict__ W,
                float* __restrict__ C, int M) {
    __shared__ float ldsW[128 * 128];   // 64 KB, row-major [K][N]
    const int lane = threadIdx.x & 31;
    const int wave = threadIdx.x >> 5;

#if GAT_HAS_TDM
    if (wave == 0) {
        // ---- Tensor DMA descriptor (ISA cdna5 §8.3/8.4), 2D: 128x128 f32 tile
        unsigned long long ga = (unsigned long long)(uintptr_t)W;
        v4u g0;
        g0.x = 1u;                                    // count=1, user mode, no gather
        g0.y = (unsigned)(uintptr_t)ldsW;             // lds_addr (bytes, WG-relative)
        g0.z = (unsigned)(ga & 0xFFFFFFFFull);        // global_addr[31:0]
        g0.w = (unsigned)((ga >> 32) & 0x01FFFFFFull) // global_addr[56:32]
             | (2u << 30);                            // type=2 ("image")
        v8i_t g1;
        g1[0] = 0x00020000;   // workgroup_mask=0, data_size=2 (4B), no pad/iterate
        g1[1] = (128 << 16);  // tensor_dim0 = 128 (low16 @ [31:16])
        g1[2] = (128 << 16);  // tensor_dim0 hi=0, tensor_dim1 = 128
        g1[3] = (128 << 16);  // tensor_dim1 hi=0, tile_dim0 = 128
        g1[4] = 128;          // tile_dim1 = 128, tile_dim2 = 0
        g1[5] = 128;          // tensor_dim0_stride = 128 (low32)
        g1[6] = 0;            // stride hi, tensor_dim1_stride lo
        g1[7] = 0;
        v4i_t z4 = {0, 0, 0, 0};
#if __clang_major__ >= 23
        v8i_t z8 = {0, 0, 0, 0, 0, 0, 0, 0};
        __builtin_amdgcn_tensor_load_to_lds(g0, g1, z4, z4, z8, 0);
#else
        __builtin_amdgcn_tensor_load_to_lds(g0, g1, z4, z4, 0);
#endif
        __builtin_amdgcn_s_wait_tensorcnt(0);
    }
    __syncthreads();
#else
    for (int i = threadIdx.x; i < 128 * 128; i += 128) ldsW[i] = W[i];
    __syncthreads();
#endif

    const int tile = blockIdx.x * 4 + wave;     // 4 waves / block
    const int tM = tile >> 3;                   // tilesN = 128/16 = 8
    const int tN = tile & 7;
    if (tM * 16 >= M) return;

    const int r  = lane & 15;   // row (A) / col (B,C) within tile
    const int kh = lane >> 4;   // K-half select for A/B fragments
    const float* Arow = A + (size_t)(tM * 16 + r) * 128;
    const float* Bcol = ldsW + tN * 16 + r;

    v8f acc = {};
    for (int k = 0; k < 128; k += 4) {
        // A 16x4 f32 frag: lanes 0-15 hold K={k,k+1}, lanes 16-31 K={k+2,k+3}
        v2f a, b;
        a.x = Arow[k + 2 * kh];
        a.y = Arow[k + 2 * kh + 1];
        // B 4x16 f32 frag from LDS (mirrored K distribution)
        b.x = Bcol[(k + 2 * kh) * 128];
        b.y = Bcol[(k + 2 * kh + 1) * 128];
        acc = __builtin_amdgcn_wmma_f32_16x16x4_f32(
            /*neg_a=*/false, a, /*neg_b=*/false, b,
            /*c_mod=*/(short)0, acc, /*reuse_a=*/false, /*reuse_b=*/false);
    }
    // C 16x16 f32: VGPR v -> row tM*16 + v + 8*kh, col tN*16 + r
    float* Crow = C + (size_t)(tM * 16 + 8 * kh) * 128 + tN * 16 + r;
#pragma unroll
    for (int v = 0; v < 8; ++v)
        Crow[(size_t)v * 128] = acc[v];
}

// ---------------- layer-1 attention scores: per (node, head), 16-wide dot
__global__ void scores_l1(const float* __restrict__ h,
                          const float* __restrict__ att_src,
                          const float* __restrict__ att_dst,
                          float* __restrict__ as, float* __restrict__ ad) {
    int i = blockIdx.x * blockDim.x + threadIdx.x;   // n*HEADS1 + hd
    if (i >= N_NODES * HEADS1) return;
    int n = i >> 3, hd = i & 7;
    const float* hp = h + (size_t)n * C1 + hd * HIDDEN;
    const float* sv = att_src + hd * HIDDEN;
    const float* dv = att_dst + hd * HIDDEN;
    float s0 = 0.f, d0 = 0.f;
#pragma unroll
    for (int c = 0; c < HIDDEN; ++c) {
        float v = hp[c];
        s0 += v * sv[c];
        d0 += v * dv[c];
    }
    as[i] = s0;
    ad[i] = d0;
}

// ---------------- layer-2 attention scores: per node, 128-wide dot (1 head)
__global__ void scores_l2(const float* __restrict__ h,
                          const float* __restrict__ att_src,
                          const float* __restrict__ att_dst,
                          float* __restrict__ as, float* __restrict__ ad) {
    int n = blockIdx.x * blockDim.x + threadIdx.x;
    if (n >= N_NODES) return;
    const float* hp = h + (size_t)n * OUT_DIM;
    float s0 = 0.f, d0 = 0.f;
#pragma unroll 8
    for (int c = 0; c < OUT_DIM; ++c) {
        float v = hp[c];
        s0 += v * att_src[c];
        d0 += v * att_dst[c];
    }
    as[n] = s0;
    ad[n] = d0;
}

// ---------------- fill
__global__ void fill_f32(float* __restrict__ p, float v, int n) {
    int i = blockIdx.x * blockDim.x + threadIdx.x;
    if (i < n) p[i] = v;
}

// ---------------- per-(edge,head): segment max of leaky_relu(scores)
template <int H>
__global__ void edge_max(const int* __restrict__ src, const int* __restrict__ dst,
                         const float* __restrict__ as, const float* __restrict__ ad,
                         float* __restrict__ nmax) {
    int i = blockIdx.x * blockDim.x + threadIdx.x;   // e*H + hd
    if (i >= N_EDGES * H) return;
    int e = i / H, hd = i % H;
    int s = src[e], d = dst[e];
    float a = as[s * H + hd] + ad[d * H + hd];
    a = (a > 0.f) ? a : a * NEG_SLOPE;
    atomicMaxF(&nmax[d * H + hd], a);
}

// ---------------- per-(edge,head): e = exp(alpha - max[dst]); denom += e
template <int H>
__global__ void edge_exp(const int* __restrict__ src, const int* __restrict__ dst,
                         const float* __restrict__ as, const float* __restrict__ ad,
                         const float* __restrict__ nmax,
                         float* __restrict__ wedge, float* __restrict__ denom) {
    int i = blockIdx.x * blockDim.x + threadIdx.x;   // e*H + hd
    if (i >= N_EDGES * H) return;
    int e = i / H, hd = i % H;
    int s = src[e], d = dst[e];
    float a = as[s * H + hd] + ad[d * H + hd];
    a = (a > 0.f) ? a : a * NEG_SLOPE;
    float w = expf(a - nmax[d * H + hd]);
    wedge[i] = w;
    atomicAdd(&denom[d * H + hd], w);
}

// ---------------- per-(edge, 4 channels): out[dst] += h[src] * alpha
// float4 gather (global_load_b128) + 4 native f32 atomic adds into L2.
template <int H>
__global__ void edge_aggr(const int* __restrict__ src, const int* __restrict__ dst,
                          const float* __restrict__ wedge, const float* __restrict__ denom,
                          const float* __restrict__ hfeat, float* __restrict__ out) {
    int i = blockIdx.x * blockDim.x + threadIdx.x;   // e*32 + quad  (< 25.6M)
    if (i >= N_EDGES * 32) return;
    int e = i >> 5, c0 = (i & 31) * 4;
    int hd = c0 / (128 / H);
    int s = src[e], d = dst[e];
    float w = wedge[e * H + hd] / (denom[d * H + hd] + GAT_EPS);
    float4 h4 = *(const float4*)(hfeat + (size_t)s * 128 + c0);
    float* op = out + (size_t)d * 128 + c0;
    atomicAdd(op + 0, h4.x * w);
    atomicAdd(op + 1, h4.y * w);
    atomicAdd(op + 2, h4.z * w);
    atomicAdd(op + 3, h4.w * w);
}

// ---------------- layer-1 epilogue: h_mid = elu(out1 + bias1), in place
__global__ void finalize_l1(float* __restrict__ out1, const float* __restrict__ bias) {
    int i = blockIdx.x * blockDim.x + threadIdx.x;
    if (i >= N_NODES * C1) return;
    float v = out1[i] + bias[i & 127];
    out1[i] = (v > 0.f) ? v : expm1f(v);
}

// ---------------- layer-2 epilogue: out += bias2 (heads=1 -> mean is identity)
__global__ void finalize_l2(float* __restrict__ out, const float* __restrict__ bias) {
    int i = blockIdx.x * blockDim.x + threadIdx.x;
    if (i >= N_NODES * OUT_DIM) return;
    out[i] = out[i] + bias[i & 127];
}

extern "C" void kernel_launch(void* const* d_in, const int* in_sizes, int n_in,
                              void* d_out, int out_size, void* d_ws, size_t ws_size,
                              hipStream_t stream) {
    const float* x        = (const float*)d_in[0];
    const int*   eidx     = (const int*)d_in[1];
    const float* W1       = (const float*)d_in[2];
    const float* att_src1 = (const float*)d_in[3];
    const float* att_dst1 = (const float*)d_in[4];
    const float* bias1    = (const float*)d_in[5];
    const float* W2       = (const float*)d_in[6];
    const float* att_src2 = (const float*)d_in[7];
    const float* att_dst2 = (const float*)d_in[8];
    const float* bias2    = (const float*)d_in[9];
    float* out = (float*)d_out;

    const int* src = eidx;              // edge_index[0]
    const int* dst = eidx + N_EDGES;    // edge_index[1]

    // ---- workspace layout (floats); total ~28.2M floats ~= 112.8 MB
    float* ws   = (float*)d_ws;
    float* h1   = ws;                    // 6,400,000  [N,128]
    float* as1  = h1  + 6400000;         //   400,000  [N,8]
    float* ad1  = as1 + 400000;          //   400,000
    float* max1 = ad1 + 400000;          //   400,000
    float* den1 = max1 + 400000;         //   400,000
    float* w1e  = den1 + 400000;         // 6,400,000  [E,8]
    float* out1 = w1e + 6400000;         // 6,400,000  [N,128] (becomes h_mid)
    float* h2   = out1 + 6400000;        // 6,400,000  [N,128]
    float* as2  = h2  + 6400000;         //    50,000
    float* ad2  = as2 + 50000;           //    50,000
    float* max2 = ad2 + 50000;           //    50,000
    float* den2 = max2 + 50000;          //    50,000
    float* w2e  = den2 + 50000;          //   800,000  [E]

    const int TPB = 256;
    // GEMM: (50000/16)*(128/16) = 25000 tiles, 4 waves (128 thr) per block
    const int gemmBlocks = 25000 / 4;

    // ================= layer 1 =================
    gemm16_wmma_f32<<<gemmBlocks, 128, 0, stream>>>(x, W1, h1, N_NODES);
    scores_l1<<<(N_NODES * HEADS1 + TPB - 1) / TPB, TPB, 0, stream>>>(h1, att_src1, att_dst1, as1, ad1);

    fill_f32<<<(N_NODES * HEADS1 + TPB - 1) / TPB, TPB, 0, stream>>>(max1, -INFINITY, N_NODES * HEADS1);
    fill_f32<<<(N_NODES * HEADS1 + TPB - 1) / TPB, TPB, 0, stream>>>(den1, 0.f, N_NODES * HEADS1);
    fill_f32<<<(N_NODES * C1 + TPB - 1) / TPB, TPB, 0, stream>>>(out1, 0.f, N_NODES * C1);

    edge_max<HEADS1><<<(N_EDGES * HEADS1 + TPB - 1) / TPB, TPB, 0, stream>>>(src, dst, as1, ad1, max1);
    edge_exp<HEADS1><<<(N_EDGES * HEADS1 + TPB - 1) / TPB, TPB, 0, stream>>>(src, dst, as1, ad1, max1, w1e, den1);
    edge_aggr<HEADS1><<<(N_EDGES * 32) / TPB, TPB, 0, stream>>>(src, dst, w1e, den1, h1, out1);
    finalize_l1<<<(N_NODES * C1 + TPB - 1) / TPB, TPB, 0, stream>>>(out1, bias1);

    // ================= layer 2 =================
    gemm16_wmma_f32<<<gemmBlocks, 128, 0, stream>>>(out1, W2, h2, N_NODES);
    scores_l2<<<(N_NODES + TPB - 1) / TPB, TPB, 0, stream>>>(h2, att_src2, att_dst2, as2, ad2);

    fill_f32<<<(N_NODES + TPB - 1) / TPB, TPB, 0, stream>>>(max2, -INFINITY, N_NODES);
    fill_f32<<<(N_NODES + TPB - 1) / TPB, TPB, 0, stream>>>(den2, 0.f, N_NODES);
    fill_f32<<<(N_NODES * OUT_DIM + TPB - 1) / TPB, TPB, 0, stream>>>(out, 0.f, N_NODES * OUT_DIM);

    edge_max<1><<<(N_EDGES + TPB - 1) / TPB, TPB, 0, stream>>>(src, dst, as2, ad2, max2);
    edge_exp<1><<<(N_EDGES + TPB - 1) / TPB, TPB, 0, stream>>>(src, dst, as2, ad2, max2, w2e, den2);
    edge_aggr<1><<<(N_EDGES * 32) / TPB, TPB, 0, stream>>>(src, dst, w2e, den2, h2, out);
    finalize_l2<<<(N_NODES * OUT_DIM + TPB - 1) / TPB, TPB, 0, stream>>>(out, bias2);
}